// _MHSA_74448963109496
// MI455X (gfx1250) — compile-verified
//
#include <hip/hip_runtime.h>

// CDNA5 / gfx1250 wave32 fp32 WMMA: D(16x16,f32) = A(16x4,f32) x B(4x16,f32) + C
// Per-lane: A,B = v2f (2 VGPRs), C/D = v8f (8 VGPRs).
typedef float v2f __attribute__((ext_vector_type(2)));
typedef float v8f __attribute__((ext_vector_type(8)));
typedef unsigned int u32x4 __attribute__((ext_vector_type(4)));
typedef int i32x4 __attribute__((ext_vector_type(4)));
typedef int i32x8 __attribute__((ext_vector_type(8)));

#define WMMA_F32(a, b, c) \
    __builtin_amdgcn_wmma_f32_16x16x4_f32(false, (a), false, (b), (short)0, (c), false, false)

#define NEG_INF (-__builtin_inff())

constexpr int Bc = 2, Tc = 4096, Dc = 512, Hc = 8, HDc = 64;
constexpr int Mc = Bc * Tc;          // 8192 rows
constexpr int QKV_N = 3 * Dc;        // 1536
constexpr size_t HEAD_ELEMS = (size_t)Bc * Hc * Tc * HDc;  // 4,194,304 per Q/K/V

// ---------------------------------------------------------------------------
// Tensor Data Mover: async 2-D f32 tile load (global -> LDS), D# per ISA ch.8.
//   g0: [w0]=count=1, [w1]=lds_addr(bytes), [w2..w3]=global_addr[56:0], type=2
//   g1: [w0]=data_size=4B | pad ctl, dims/tiles/stride packed per spec
// pad_on: +1 DWORD after every 64 DWORDs (pad_interval=5, pad_amount=0)
//         -> reproduces a [rows][65] float LDS layout for conflict-free reads.
// Tracked on TENSORcnt; completes in order with other tensor ops of the wave.
// ---------------------------------------------------------------------------
__device__ __forceinline__ void tdm_load_tile_f32(
    const float* gptr, unsigned lds_byte_off,
    unsigned tensor_d0, unsigned tensor_d1,
    unsigned tile_d0, unsigned tile_d1,
    unsigned d0_stride, bool pad_on) {
  unsigned long long ga = (unsigned long long)(const void*)gptr;
  unsigned w0 = (2u << 16);                       // data_size = 4 bytes
  if (pad_on) w0 |= (1u << 20) | (5u << 22);      // pad_enable, interval=64 DW, amount=1 DW
  u32x4 g0 = { 1u,                                // count=1 (valid descriptor)
               lds_byte_off,
               (unsigned)ga,
               (unsigned)((ga >> 32) & 0x1FFFFFFu) | (2u << 30) };  // type=2
  i32x8 g1 = { (int)w0,
               (int)((tensor_d0 & 0xFFFFu) << 16),                         // td0[15:0] @63:48
               (int)(((tensor_d0 >> 16) & 0xFFFFu) |
                     ((tensor_d1 & 0xFFFFu) << 16)),                       // td0 hi, td1 lo
               (int)(((tensor_d1 >> 16) & 0xFFFFu) | (tile_d0 << 16)),     // td1 hi, tile0
               (int)(tile_d1 & 0xFFFFu),                                   // tile1 (tile2=0)
               (int)d0_stride,                                             // dim0 stride lo32
               0, 0 };
  i32x4 z4 = {0, 0, 0, 0};
#if defined(__clang_major__) && __clang_major__ >= 23
  i32x8 z8 = {0, 0, 0, 0, 0, 0, 0, 0};
  __builtin_amdgcn_tensor_load_to_lds(g0, g1, z4, z4, z8, 0);
#else
  __builtin_amdgcn_tensor_load_to_lds(g0, g1, z4, z4, 0);
#endif
}

// ---------------------------------------------------------------------------
// Kernel 1: qkv = x @ qkv_w + qkv_b, scattered to Q/K/V in [B,H,T,64] layout.
// 64x64 tile per block, 4 waves, each wave: 16 rows x 64 cols (4 v8f accums).
// ---------------------------------------------------------------------------
__global__ __launch_bounds__(128) void qkv_gemm_kernel(
    const float* __restrict__ x, const float* __restrict__ w,
    const float* __restrict__ bias,
    float* __restrict__ Q, float* __restrict__ K, float* __restrict__ V) {
  __shared__ float Xs[64][33];   // 64 rows x 32 K, pad -> conflict-free A reads
  __shared__ float Ws[32][64];   // 32 K x 64 cols, row-uniform B reads
  const int tid  = threadIdx.x;
  const int wave = tid >> 5;
  const int lane = tid & 31;
  const int l15  = lane & 15;
  const int half = lane >> 4;
  const int m0 = blockIdx.x * 64;
  const int n0 = blockIdx.y * 64;

  v8f acc[4] = {};

  for (int k0 = 0; k0 < Dc; k0 += 32) {
    for (int e = tid; e < 64 * 32; e += 128) {
      int r = e >> 5, c = e & 31;
      Xs[r][c] = x[(size_t)(m0 + r) * Dc + k0 + c];
    }
    for (int e = tid; e < 32 * 64; e += 128) {
      int r = e >> 6, c = e & 63;
      Ws[r][c] = w[(size_t)(k0 + r) * QKV_N + n0 + c];
    }
    __syncthreads();
#pragma unroll
    for (int kk = 0; kk < 32; kk += 4) {
      v2f a;
      a.x = Xs[wave * 16 + l15][kk + 2 * half];
      a.y = Xs[wave * 16 + l15][kk + 2 * half + 1];
#pragma unroll
      for (int j = 0; j < 4; ++j) {
        v2f b;
        b.x = Ws[kk + 2 * half][j * 16 + l15];
        b.y = Ws[kk + 2 * half + 1][j * 16 + l15];
        acc[j] = WMMA_F32(a, b, acc[j]);
      }
    }
    __syncthreads();
  }

  // Epilogue: bias + scatter. n in [0,512)->Q, [512,1024)->K, [1024,1536)->V.
#pragma unroll
  for (int j = 0; j < 4; ++j) {
    int n = n0 + j * 16 + l15;
    float bv = bias[n];
    int seg = n >> 9;
    int d = n & 511;
    int h = d >> 6, hd = d & 63;
    float* dst = (seg == 0) ? Q : (seg == 1) ? K : V;
#pragma unroll
    for (int i = 0; i < 8; ++i) {
      int m = m0 + wave * 16 + i + 8 * half;
      int b = m >> 12;       // / T (4096)
      int t = m & 4095;
      dst[(((size_t)(b * Hc + h) * Tc + t) << 6) + hd] = acc[j][i] + bv;
    }
  }
}

// ---------------------------------------------------------------------------
// Kernel 2: causal flash attention for one (b,h) and one 64-row query block.
// 4 waves; wave w owns query rows [q0+16w, q0+16w+16). Online softmax.
// K/V tiles are DMA'd by the Tensor Data Mover with double buffering: wave 0
// issues the next block's tensor_load_to_lds before computing on the current
// one, so HBM latency overlaps the ~128 WMMAs + softmax per block. The TDM
// pads K rows to stride 65 in-flight for conflict-free K^T fragment reads.
// P (accumulator layout -> A-fragment layout) round-trips through the CURRENT
// K buffer, which is safe: the prefetch always targets the other buffer.
// ---------------------------------------------------------------------------
__global__ __launch_bounds__(128) void attn_kernel(
    const float* __restrict__ Q, const float* __restrict__ K,
    const float* __restrict__ V, float* __restrict__ O) {
  __shared__ __align__(16) float Qs[64][65];
  __shared__ __align__(16) float Ks[2][64][65];  // TDM-padded; reused for P
  __shared__ __align__(16) float Vs[2][64][64];
  const int tid  = threadIdx.x;
  const int wave = tid >> 5;
  const int lane = tid & 31;
  const int l15  = lane & 15;
  const int half = lane >> 4;
  const int qb = blockIdx.x;
  const int bh = blockIdx.y;
  const int q0 = qb * 64;

  const float* Qbh = Q + (size_t)bh * Tc * HDc;
  const float* Kbh = K + (size_t)bh * Tc * HDc;
  const float* Vbh = V + (size_t)bh * Tc * HDc;

  // Stage Q once, pre-scaled by 1/sqrt(HD) = 0.125.
  for (int e = tid; e < 64 * 64; e += 128) {
    int r = e >> 6, c = e & 63;
    Qs[r][c] = Qbh[(size_t)(q0 + r) * HDc + c] * 0.125f;
  }

  v8f o_acc[4] = {};
  float m_i[8], l_i[8];
#pragma unroll
  for (int i = 0; i < 8; ++i) { m_i[i] = NEG_INF; l_i[i] = 0.f; }

  // LDS byte offsets of the ping-pong buffers (generic LDS addr low 32 bits).
  const unsigned ks_off[2] = { (unsigned)(size_t)&Ks[0][0][0],
                               (unsigned)(size_t)&Ks[1][0][0] };
  const unsigned vs_off[2] = { (unsigned)(size_t)&Vs[0][0][0],
                               (unsigned)(size_t)&Vs[1][0][0] };

  // Prefetch block 0 (2 tensor ops per batch; in-order completion).
  if (wave == 0) {
    tdm_load_tile_f32(Kbh, ks_off[0], HDc, Tc, 64, 64, HDc, true);
    tdm_load_tile_f32(Vbh, vs_off[0], HDc, Tc, 64, 64, HDc, false);
  }

  for (int kb = 0; kb <= qb; ++kb) {  // causal: skip blocks above the diagonal
    const int cur = kb & 1;
    const int k0 = kb * 64;
    if (wave == 0) {
      if (kb + 1 <= qb) {
        const float* Kn = Kbh + (size_t)(k0 + 64) * HDc;
        const float* Vn = Vbh + (size_t)(k0 + 64) * HDc;
        tdm_load_tile_f32(Kn, ks_off[cur ^ 1], HDc, Tc, 64, 64, HDc, true);
        tdm_load_tile_f32(Vn, vs_off[cur ^ 1], HDc, Tc, 64, 64, HDc, false);
        __builtin_amdgcn_s_wait_tensorcnt(2);  // batch(kb) done; batch(kb+1) in flight
      } else {
        __builtin_amdgcn_s_wait_tensorcnt(0);
      }
    }
    __syncthreads();   // K/V(cur) visible to all waves (also fences Qs writes)

    float (*Ps)[65] = &Ks[cur][wave * 16];   // per-wave 16x64 P slice (later)

    // S(16x64) = Qslice(16x64) x K(64x64)^T
    v8f s[4] = {};
#pragma unroll
    for (int kd = 0; kd < HDc; kd += 4) {
      v2f a;
      a.x = Qs[wave * 16 + l15][kd + 2 * half];
      a.y = Qs[wave * 16 + l15][kd + 2 * half + 1];
#pragma unroll
      for (int j = 0; j < 4; ++j) {
        v2f b;                                   // B[kd][key] = K[key][kd]
        b.x = Ks[cur][j * 16 + l15][kd + 2 * half];
        b.y = Ks[cur][j * 16 + l15][kd + 2 * half + 1];
        s[j] = WMMA_F32(a, b, s[j]);
      }
    }

    if (kb == qb) {   // mask only on the diagonal block
#pragma unroll
      for (int j = 0; j < 4; ++j)
#pragma unroll
        for (int i = 0; i < 8; ++i) {
          int row = q0 + wave * 16 + i + 8 * half;
          int col = k0 + j * 16 + l15;
          if (col > row) s[j][i] = NEG_INF;
        }
    }

    // Online softmax. Row (i + 8*half) lives in 16 lanes of one half-wave,
    // so xor-shuffles 8/4/2/1 reduce within the correct lane group.
#pragma unroll
    for (int i = 0; i < 8; ++i) {
      float rm = fmaxf(fmaxf(s[0][i], s[1][i]), fmaxf(s[2][i], s[3][i]));
      rm = fmaxf(rm, __shfl_xor(rm, 8, 32));
      rm = fmaxf(rm, __shfl_xor(rm, 4, 32));
      rm = fmaxf(rm, __shfl_xor(rm, 2, 32));
      rm = fmaxf(rm, __shfl_xor(rm, 1, 32));
      float m_new = fmaxf(m_i[i], rm);
      float alpha = __expf(m_i[i] - m_new);
      float rs = 0.f;
#pragma unroll
      for (int j = 0; j < 4; ++j) {
        float p = __expf(s[j][i] - m_new);
        s[j][i] = p;
        rs += p;
      }
      rs += __shfl_xor(rs, 8, 32);
      rs += __shfl_xor(rs, 4, 32);
      rs += __shfl_xor(rs, 2, 32);
      rs += __shfl_xor(rs, 1, 32);
      l_i[i] = l_i[i] * alpha + rs;
      m_i[i] = m_new;
#pragma unroll
      for (int j = 0; j < 4; ++j) o_acc[j][i] *= alpha;  // rescale O (C input)
    }

    __syncthreads();   // everyone done reading K(cur) before we overwrite as P
#pragma unroll
    for (int j = 0; j < 4; ++j)
#pragma unroll
      for (int i = 0; i < 8; ++i)
        Ps[i + 8 * half][j * 16 + l15] = s[j][i];
    __syncthreads();

    // O(16x64) += P(16x64) x V(64x64)
#pragma unroll
    for (int kk = 0; kk < 64; kk += 4) {
      v2f a;
      a.x = Ps[l15][kk + 2 * half];
      a.y = Ps[l15][kk + 2 * half + 1];
#pragma unroll
      for (int j = 0; j < 4; ++j) {
        v2f b;
        b.x = Vs[cur][kk + 2 * half][j * 16 + l15];
        b.y = Vs[cur][kk + 2 * half + 1][j * 16 + l15];
        o_acc[j] = WMMA_F32(a, b, o_acc[j]);
      }
    }
    __syncthreads();   // protect P(cur)/V(cur) before next iteration reuse
  }

  // Normalize and store to [B,T,D] with d = h*64 + hd (head-merge layout).
  int b = bh >> 3, h = bh & 7;
#pragma unroll
  for (int i = 0; i < 8; ++i) {
    float inv = 1.f / l_i[i];
    int t = q0 + wave * 16 + i + 8 * half;
    size_t base = ((size_t)(b * Tc + t)) * Dc + h * HDc;
#pragma unroll
    for (int j = 0; j < 4; ++j)
      O[base + j * 16 + l15] = o_acc[j][i] * inv;
  }
}

// ---------------------------------------------------------------------------
// Kernel 3: out = att @ proj_w + proj_b   ([8192,512] x [512,512])
// ---------------------------------------------------------------------------
__global__ __launch_bounds__(128) void proj_gemm_kernel(
    const float* __restrict__ a_in, const float* __restrict__ w,
    const float* __restrict__ bias, float* __restrict__ out) {
  __shared__ float Xs[64][33];
  __shared__ float Ws[32][64];
  const int tid  = threadIdx.x;
  const int wave = tid >> 5;
  const int lane = tid & 31;
  const int l15  = lane & 15;
  const int half = lane >> 4;
  const int m0 = blockIdx.x * 64;
  const int n0 = blockIdx.y * 64;

  v8f acc[4] = {};

  for (int k0 = 0; k0 < Dc; k0 += 32) {
    for (int e = tid; e < 64 * 32; e += 128) {
      int r = e >> 5, c = e & 31;
      Xs[r][c] = a_in[(size_t)(m0 + r) * Dc + k0 + c];
    }
    for (int e = tid; e < 32 * 64; e += 128) {
      int r = e >> 6, c = e & 63;
      Ws[r][c] = w[(size_t)(k0 + r) * Dc + n0 + c];
    }
    __syncthreads();
#pragma unroll
    for (int kk = 0; kk < 32; kk += 4) {
      v2f a;
      a.x = Xs[wave * 16 + l15][kk + 2 * half];
      a.y = Xs[wave * 16 + l15][kk + 2 * half + 1];
#pragma unroll
      for (int j = 0; j < 4; ++j) {
        v2f b;
        b.x = Ws[kk + 2 * half][j * 16 + l15];
        b.y = Ws[kk + 2 * half + 1][j * 16 + l15];
        acc[j] = WMMA_F32(a, b, acc[j]);
      }
    }
    __syncthreads();
  }

#pragma unroll
  for (int j = 0; j < 4; ++j) {
    int n = n0 + j * 16 + l15;
    float bv = bias[n];
#pragma unroll
    for (int i = 0; i < 8; ++i) {
      int m = m0 + wave * 16 + i + 8 * half;
      out[(size_t)m * Dc + n] = acc[j][i] + bv;
    }
  }
}

// ---------------------------------------------------------------------------
// Launcher. Inputs (setup_inputs order): x, mask, qkv_w, qkv_b, proj_w, proj_b.
// mask is implicit (causal) — unused. Workspace: Q | K | V | att (4 x 16 MB).
// ---------------------------------------------------------------------------
extern "C" void kernel_launch(void* const* d_in, const int* in_sizes, int n_in,
                              void* d_out, int out_size, void* d_ws, size_t ws_size,
                              hipStream_t stream) {
  const float* x      = (const float*)d_in[0];
  // d_in[1]: causal mask (int32) — causality is hardcoded in attn_kernel.
  const float* qkv_w  = (const float*)d_in[2];
  const float* qkv_b  = (const float*)d_in[3];
  const float* proj_w = (const float*)d_in[4];
  const float* proj_b = (const float*)d_in[5];
  float* out = (float*)d_out;

  float* ws  = (float*)d_ws;
  float* Qw  = ws;
  float* Kw  = ws + HEAD_ELEMS;
  float* Vw  = ws + 2 * HEAD_ELEMS;
  float* att = ws + 3 * HEAD_ELEMS;

  dim3 g1(Mc / 64, QKV_N / 64);         // 128 x 24
  qkv_gemm_kernel<<<g1, 128, 0, stream>>>(x, qkv_w, qkv_b, Qw, Kw, Vw);

  dim3 g2(Tc / 64, Bc * Hc);            // 64 x 16
  attn_kernel<<<g2, 128, 0, stream>>>(Qw, Kw, Vw, att);

  dim3 g3(Mc / 64, Dc / 64);            // 128 x 8
  proj_gemm_kernel<<<g3, 128, 0, stream>>>(att, proj_w, proj_b, out);
}